// VisionEncoderMambaBlock_85736137162950
// MI455X (gfx1250) — compile-verified
//
#include <hip/hip_runtime.h>
#include <math.h>

#define DIMC     256
#define DT_RANK  16
#define D_STATE  16
#define BATCH    8
#define SEQL     2048
#define M_TOT    (BATCH * SEQL)     /* 16384 rows */
#define LN_EPS   1e-5f
#define TCHUNK   64

typedef __attribute__((ext_vector_type(2))) float v2f;
typedef __attribute__((ext_vector_type(8))) float v8f;

// ---------------------------------------------------------------------------
// Activation helpers
// ---------------------------------------------------------------------------
__device__ __forceinline__ float act_silu(float v) {
    return v / (1.f + __expf(-v));
}
__device__ __forceinline__ float act_softplus(float v) {
    return (v > 20.f) ? v : log1pf(__expf(v));
}

// fp32 WMMA fragment loader: row-major tile base, `ldm` floats per row.
// lane 0-15  -> row = lane,    cols k..k+1
// lane 16-31 -> row = lane-16, cols k+2..k+3   (ISA 7.12.2, 32-bit A 16x4)
__device__ __forceinline__ v2f ld_frag(const float* base, int ldm, int lane, int k) {
    int r  = lane & 15;
    int kk = k + ((lane >> 4) << 1);
    v2f f;
    f.x = base[r * ldm + kk];
    f.y = base[r * ldm + kk + 1];
    return f;
}

// ---------------------------------------------------------------------------
// Async global -> LDS copy (CDNA5, ASYNCcnt-tracked). 16B per lane.
// ---------------------------------------------------------------------------
__device__ __forceinline__ unsigned lds_off_u32(const void* p) {
    return (unsigned)(uintptr_t)(const __attribute__((address_space(3))) void*)p;
}
__device__ __forceinline__ void async_copy_b128(const void* lds_dst, const float* gsrc) {
    unsigned           l = lds_off_u32(lds_dst);
    unsigned long long g = (unsigned long long)(uintptr_t)gsrc;
    asm volatile("global_load_async_to_lds_b128 %0, %1, off"
                 :: "v"(l), "v"(g) : "memory");
}

// ---------------------------------------------------------------------------
// Kernel 1: LayerNorm. One wave per row (8 rows / 256-thread block).
// ---------------------------------------------------------------------------
__global__ __launch_bounds__(256) void ln_kernel(
    const float* __restrict__ x, const float* __restrict__ w,
    const float* __restrict__ b, float* __restrict__ xn) {
    int row  = blockIdx.x * 8 + (threadIdx.x >> 5);
    int lane = threadIdx.x & 31;
    const float* xr = x + (size_t)row * DIMC;

    float v[8];
    float s = 0.f;
#pragma unroll
    for (int i = 0; i < 8; ++i) { v[i] = xr[lane + 32 * i]; s += v[i]; }
#pragma unroll
    for (int off = 16; off >= 1; off >>= 1) s += __shfl_xor(s, off, 32);
    float mu = s * (1.f / DIMC);

    float var = 0.f;
#pragma unroll
    for (int i = 0; i < 8; ++i) { float d = v[i] - mu; var += d * d; }
#pragma unroll
    for (int off = 16; off >= 1; off >>= 1) var += __shfl_xor(var, off, 32);
    float inv = rsqrtf(var * (1.f / DIMC) + LN_EPS);

    float* xo = xn + (size_t)row * DIMC;
#pragma unroll
    for (int i = 0; i < 8; ++i) {
        int c = lane + 32 * i;
        xo[c] = (v[i] - mu) * inv * w[c] + b[c];
    }
}

// ---------------------------------------------------------------------------
// Kernel 2: C[M x 256] = act(A[M x 256] @ W[256 x 256]^T + bias)
// WG: 4 waves, 16 rows x 64 cols. LDS: A tile 16x256 + W panel 64x256.
// K-loop of V_WMMA_F32_16X16X4_F32.
// ---------------------------------------------------------------------------
__global__ __launch_bounds__(128) void gemm256_kernel(
    const float* __restrict__ A, const float* __restrict__ W,
    const float* __restrict__ bias, float* __restrict__ C, int act) {
    __shared__ float Al[16 * DIMC];     // 16 KB
    __shared__ float Wl[64 * DIMC];     // 64 KB
    int m0   = blockIdx.x * 16;
    int n0wg = blockIdx.y * 64;
    int tid  = threadIdx.x;
    int wave = tid >> 5, lane = tid & 31;

    const float4* Ag = (const float4*)(A + (size_t)m0 * DIMC);
    float4* Al4 = (float4*)Al;
#pragma unroll
    for (int i = tid; i < 16 * DIMC / 4; i += 128) Al4[i] = Ag[i];
    const float4* Wg = (const float4*)(W + (size_t)n0wg * DIMC);
    float4* Wl4 = (float4*)Wl;
#pragma unroll 4
    for (int i = tid; i < 64 * DIMC / 4; i += 128) Wl4[i] = Wg[i];
    __syncthreads();

    const float* Wp = Wl + wave * 16 * DIMC;
    v8f acc = {};
#pragma unroll 8
    for (int k = 0; k < DIMC; k += 4) {
        v2f a = ld_frag(Al, DIMC, lane, k);
        v2f b = ld_frag(Wp, DIMC, lane, k);
        acc = __builtin_amdgcn_wmma_f32_16x16x4_f32(
            false, a, false, b, (short)0, acc, false, false);
    }

    int n0    = n0wg + wave * 16;
    int col   = lane & 15;
    int rbase = (lane >> 4) * 8;
    float bv  = bias[n0 + col];
#pragma unroll
    for (int v = 0; v < 8; ++v) {
        float val = acc[v] + bv;
        val = (act == 0) ? act_silu(val) : act_softplus(val);
        C[(size_t)(m0 + rbase + v) * DIMC + n0 + col] = val;
    }
}

// ---------------------------------------------------------------------------
// Kernel 3: fused dbc (48-wide) + delta = softplus(dbc[:, :16] @ dt_W^T + b)
// WG: 16 waves, 16 rows. Phase 1: waves 0-2 compute dbc tiles (K=256) into
// LDS; waves 1/2 also emit raw B/C. Phase 2: all 16 waves compute the
// 256-wide delta with K=16.
// ---------------------------------------------------------------------------
__global__ __launch_bounds__(512) void dbcdt_kernel(
    const float* __restrict__ h, const float* __restrict__ dbcW,   // 48 x 256
    const float* __restrict__ dtW,                                 // 256 x 16
    const float* __restrict__ dtb,
    float* __restrict__ Bm, float* __restrict__ Cm,                // M x 16
    float* __restrict__ delta) {                                   // M x 256
    __shared__ float Hl[16 * DIMC];     // 16 KB
    __shared__ float Dbc[16 * 48];      // 3 KB (raw dbc tile)
    int m0   = blockIdx.x * 16;
    int tid  = threadIdx.x;
    int wave = tid >> 5, lane = tid & 31;

    const float4* Hg = (const float4*)(h + (size_t)m0 * DIMC);
    float4* Hl4 = (float4*)Hl;
    for (int i = tid; i < 16 * DIMC / 4; i += 512) Hl4[i] = Hg[i];
    __syncthreads();

    int col   = lane & 15;
    int rbase = (lane >> 4) * 8;

    if (wave < 3) {                       // wave-uniform: EXEC stays all-1s
        int n0 = wave * 16;
        v8f acc = {};
#pragma unroll 8
        for (int k = 0; k < DIMC; k += 4) {
            v2f a = ld_frag(Hl, DIMC, lane, k);
            v2f b = ld_frag(dbcW + (size_t)n0 * DIMC, DIMC, lane, k);
            acc = __builtin_amdgcn_wmma_f32_16x16x4_f32(
                false, a, false, b, (short)0, acc, false, false);
        }
#pragma unroll
        for (int v = 0; v < 8; ++v) {
            float val = acc[v];
            Dbc[(rbase + v) * 48 + n0 + col] = val;
            if (wave == 1) Bm[(size_t)(m0 + rbase + v) * D_STATE + col] = val;
            if (wave == 2) Cm[(size_t)(m0 + rbase + v) * D_STATE + col] = val;
        }
    }
    __syncthreads();

    int e0 = wave * 16;
    v8f acc2 = {};
#pragma unroll
    for (int k = 0; k < DT_RANK; k += 4) {
        v2f a = ld_frag(Dbc, 48, lane, k);                    // delta_raw cols 0..15
        v2f b = ld_frag(dtW + (size_t)e0 * DT_RANK, DT_RANK, lane, k);
        acc2 = __builtin_amdgcn_wmma_f32_16x16x4_f32(
            false, a, false, b, (short)0, acc2, false, false);
    }
    float bv = dtb[e0 + col];
#pragma unroll
    for (int v = 0; v < 8; ++v) {
        float val = act_softplus(acc2[v] + bv);
        delta[(size_t)(m0 + rbase + v) * DIMC + e0 + col] = val;
    }
}

// ---------------------------------------------------------------------------
// Kernel 4: selective scan. Block = (batch b, 16-channel block e0).
// lane -> (channel = wave*2 + lane/16, state n = lane&15).
// Double-buffered chunks of TCHUNK timesteps staged with
// GLOBAL_LOAD_ASYNC_TO_LDS_B128 (ASYNCcnt), overlapping the next chunk's
// HBM traffic with the current chunk's serial recurrence.
// Each wave issues exactly 4 async loads per chunk -> s_wait_asynccnt 4
// guarantees the previous chunk landed.
// ---------------------------------------------------------------------------
__global__ __launch_bounds__(256) void scan_kernel(
    const float* __restrict__ h, const float* __restrict__ delta,
    const float* __restrict__ Bm, const float* __restrict__ Cm,
    const float* __restrict__ A_log, const float* __restrict__ Dvec,
    float* __restrict__ y) {
    int b    = blockIdx.x >> 4;
    int e0   = (blockIdx.x & 15) * 16;
    int tid  = threadIdx.x;
    int wave = tid >> 5, lane = tid & 31;
    int cw = wave * 2 + (lane >> 4);     // channel within block: 0..15
    int c  = e0 + cw;
    int n  = lane & 15;

    float Aen = -__expf(A_log[c * D_STATE + n]);
    float Dv  = Dvec[c];

    __shared__ float Bc[2][TCHUNK * 16];
    __shared__ float Cc[2][TCHUNK * 16];
    __shared__ float dc[2][TCHUNK * 16];
    __shared__ float hc[2][TCHUNK * 16];
    __shared__ float yc[TCHUNK * 16];

    const size_t rowbase = (size_t)b * SEQL;

    // Per-thread async transfer geometry: 256 threads x 16B per array.
    int ti = tid;                 // B/C: contiguous 4KB region, chunk ti
    int tr = tid >> 2;            // delta/h: row within chunk
    int c4 = (tid & 3) * 4;       // delta/h: 16B sub-segment of 64B row

    auto issue_chunk = [&](int t0, int sel) {
        size_t r0 = rowbase + t0;
        async_copy_b128(&Bc[sel][ti * 4], Bm + r0 * D_STATE + ti * 4);
        async_copy_b128(&Cc[sel][ti * 4], Cm + r0 * D_STATE + ti * 4);
        async_copy_b128(&dc[sel][tr * 16 + c4], delta + (r0 + tr) * DIMC + e0 + c4);
        async_copy_b128(&hc[sel][tr * 16 + c4], h + (r0 + tr) * DIMC + e0 + c4);
    };

    issue_chunk(0, 0);                    // prologue: 4 async loads in flight

    float s = 0.f;
    int sel = 0;
    for (int t0 = 0; t0 < SEQL; t0 += TCHUNK, sel ^= 1) {
        if (t0 + TCHUNK < SEQL) {
            issue_chunk(t0 + TCHUNK, sel ^ 1);            // 8 in flight
            asm volatile("s_wait_asynccnt 0x4" ::: "memory");  // oldest 4 done
        } else {
            asm volatile("s_wait_asynccnt 0x0" ::: "memory");
        }
        __syncthreads();

        const float* Bp = Bc[sel];
        const float* Cp = Cc[sel];
        const float* dp = dc[sel];
        const float* hp = hc[sel];
#pragma unroll 4
        for (int tt = 0; tt < TCHUNK; ++tt) {
            float dlt = dp[tt * 16 + cw];
            float xv  = hp[tt * 16 + cw];
            float dA  = __expf(dlt * Aen);
            s = dA * s + dlt * Bp[tt * 16 + n] * xv;
            float p = s * Cp[tt * 16 + n];
            p += __shfl_xor(p, 8, 16);
            p += __shfl_xor(p, 4, 16);
            p += __shfl_xor(p, 2, 16);
            p += __shfl_xor(p, 1, 16);
            if (n == 0) yc[tt * 16 + cw] = p + Dv * xv;
        }
        __syncthreads();

        for (int i = tid; i < TCHUNK * 16; i += 256) {
            int tt = i >> 4, col = i & 15;
            y[(rowbase + t0 + tt) * DIMC + e0 + col] = yc[i];
        }
        __syncthreads();
    }
}

// ---------------------------------------------------------------------------
// Kernel 5: out = (yf + yb) * z + x   (float4 elementwise)
// ---------------------------------------------------------------------------
__global__ __launch_bounds__(256) void combine_kernel(
    const float* __restrict__ x, const float* __restrict__ z,
    const float* __restrict__ yf, const float* __restrict__ yb,
    float* __restrict__ out) {
    size_t i = (size_t)blockIdx.x * 256 + threadIdx.x;
    float4 xv = ((const float4*)x)[i];
    float4 zv = ((const float4*)z)[i];
    float4 fv = ((const float4*)yf)[i];
    float4 bv = ((const float4*)yb)[i];
    float4 o;
    o.x = (fv.x + bv.x) * zv.x + xv.x;
    o.y = (fv.y + bv.y) * zv.y + xv.y;
    o.z = (fv.z + bv.z) * zv.z + xv.z;
    o.w = (fv.w + bv.w) * zv.w + xv.w;
    ((float4*)out)[i] = o;
}

// ---------------------------------------------------------------------------
extern "C" void kernel_launch(void* const* d_in, const int* in_sizes, int n_in,
                              void* d_out, int out_size, void* d_ws, size_t ws_size,
                              hipStream_t stream) {
    const float* x      = (const float*)d_in[0];
    const float* ln_w   = (const float*)d_in[1];
    const float* ln_b   = (const float*)d_in[2];
    const float* proj_W = (const float*)d_in[3];
    const float* proj_b = (const float*)d_in[4];
    const float* fwd_W  = (const float*)d_in[5];
    const float* fwd_b  = (const float*)d_in[6];
    const float* bwd_W  = (const float*)d_in[7];
    const float* bwd_b  = (const float*)d_in[8];
    const float* dbc_W  = (const float*)d_in[9];
    const float* dt_W   = (const float*)d_in[10];
    const float* dt_b   = (const float*)d_in[11];
    const float* A_log  = (const float*)d_in[12];
    const float* Dvec   = (const float*)d_in[13];
    float* out = (float*)d_out;

    float* ws = (float*)d_ws;
    const size_t NE = (size_t)M_TOT * DIMC;       // 4,194,304
    float* xn  = ws;
    float* z   = ws + 1 * NE;
    float* hf  = ws + 2 * NE;
    float* hb  = ws + 3 * NE;
    float* df  = ws + 4 * NE;
    float* db  = ws + 5 * NE;
    float* yf  = ws + 6 * NE;
    float* yb  = ws + 7 * NE;
    float* Bf  = ws + 8 * NE;
    float* Cf  = Bf + (size_t)M_TOT * D_STATE;
    float* Bb  = Cf + (size_t)M_TOT * D_STATE;
    float* Cb  = Bb + (size_t)M_TOT * D_STATE;

    // 1) LayerNorm
    ln_kernel<<<M_TOT / 8, 256, 0, stream>>>(x, ln_w, ln_b, xn);

    // 2) Three WMMA GEMMs: z = silu(xn@projW^T+b), h = softplus(xn@W^T+b)
    dim3 ggrid(M_TOT / 16, DIMC / 64);
    gemm256_kernel<<<ggrid, 128, 0, stream>>>(xn, proj_W, proj_b, z, 0);
    gemm256_kernel<<<ggrid, 128, 0, stream>>>(xn, fwd_W, fwd_b, hf, 1);
    gemm256_kernel<<<ggrid, 128, 0, stream>>>(xn, bwd_W, bwd_b, hb, 1);

    // 3) dbc + dt per direction
    dbcdt_kernel<<<M_TOT / 16, 512, 0, stream>>>(hf, dbc_W, dt_W, dt_b, Bf, Cf, df);
    dbcdt_kernel<<<M_TOT / 16, 512, 0, stream>>>(hb, dbc_W, dt_W, dt_b, Bb, Cb, db);

    // 4) Selective scan per direction (async double-buffered)
    scan_kernel<<<BATCH * 16, 256, 0, stream>>>(hf, df, Bf, Cf, A_log, Dvec, yf);
    scan_kernel<<<BATCH * 16, 256, 0, stream>>>(hb, db, Bb, Cb, A_log, Dvec, yb);

    // 5) Combine
    combine_kernel<<<(int)(NE / 4 / 256), 256, 0, stream>>>(x, z, yf, yb, out);
}